// FrequencyMatchedVQC_2104533975409
// MI455X (gfx1250) — compile-verified
//
#include <hip/hip_runtime.h>

// ---------------------------------------------------------------------------
// Frequency-matched VQC, 14 qubits, depth 6, B=2048.
// One workgroup (256 threads, 8 wave32) per batch sample.
// State (2^14 complex) double-buffered in LDS; variational RY layers applied
// as real 16x16 Kronecker matrices via chained V_WMMA_F32_16X16X4_F32.
// CNOT ladder folded analytically into a Gray-code gather i ^ (i>>1), fused
// into the first WMMA pass's read addresses each layer.
// ---------------------------------------------------------------------------

typedef float v2f __attribute__((ext_vector_type(2)));
typedef float v8f __attribute__((ext_vector_type(8)));

#define NQ     14
#define DIM    16384      // 2^14 amplitudes
#define VDEPTH 6
#define BLOCK  256
#define RSQRT2 0.70710678118654752440f

// LDS layout (floats). Two full complex state buffers (ping-pong) + scratch.
#define OFF_RE0 0
#define OFF_IM0 16384
#define OFF_RE1 32768
#define OFF_IM1 49152
#define OFF_M   65536              // 3 x (16x16) layer matrices = 768 floats
#define OFF_M3  (OFF_M + 768)      // 4x4 matrix for wires 12,13
#define OFF_CS  (OFF_M3 + 16)     // cos/sin of layer params, 28 floats
#define OFF_VT  (OFF_CS + 28)     // encoding per-wire 2-vectors, 14*4 floats
#define LDS_FLOATS (OFF_VT + 56)

static_assert((size_t)LDS_FLOATS * 4 <= 320u * 1024u, "exceeds 320KB WGP LDS");

// Flat state index for group g: fiber row j (4 wires), column c in [0,1024).
// g=0 -> wires 0-3  (bits 13..10), g=1 -> wires 4-7 (bits 9..6),
// g=2 -> wires 8-11 (bits 5..2).
template <int G>
__device__ __forceinline__ int groupAddr(int j, int c) {
  if constexpr (G == 0) return (j << 10) | c;
  else if constexpr (G == 1) return ((c >> 6) << 10) | (j << 6) | (c & 63);
  else return ((c >> 2) << 6) | (j << 2) | (c & 3);
}

// D(16x1024) = M(16x16) * S(16x1024), applied to re and im planes.
// PERM fuses the CNOT-ladder Gray-code gather into the source addresses.
// The tile loop has a compile-time trip count of 8 (tiles (i<<3)|wid), so the
// wave loop is exec-uniform: EXEC stays all-ones around every WMMA.
template <int G, bool PERM>
__device__ __forceinline__ void apply_group16(
    const float* __restrict__ sRe, const float* __restrict__ sIm,
    float* __restrict__ dRe, float* __restrict__ dIm,
    const float* __restrict__ M)
{
  const int lane = threadIdx.x & 31;
  const int wid  = (threadIdx.x >> 5) & 7;
  const int m    = lane & 15;
  const int koff = (lane >> 4) << 1;   // 0 for lanes 0-15, 2 for lanes 16-31

  // A-matrix (16x4 chunks of M): vgpr.x holds K=koff, vgpr.y holds K=koff+1.
  v2f a0, a1, a2, a3;
  a0.x = M[m * 16 +  0 + koff]; a0.y = M[m * 16 +  1 + koff];
  a1.x = M[m * 16 +  4 + koff]; a1.y = M[m * 16 +  5 + koff];
  a2.x = M[m * 16 +  8 + koff]; a2.y = M[m * 16 +  9 + koff];
  a3.x = M[m * 16 + 12 + koff]; a3.y = M[m * 16 + 13 + koff];

  const int rbase = (lane >> 4) << 3;  // D rows: v + 0 or v + 8

#pragma unroll 2
  for (int i = 0; i < 8; ++i) {        // 64 column tiles / 8 waves, uniform
    const int t   = (i << 3) | wid;
    const int col = (t << 4) | m;
    v2f bRe[4], bIm[4];
#pragma unroll
    for (int k = 0; k < 4; ++k) {
      const int r0 = (k << 2) + koff;
      int ad0 = groupAddr<G>(r0,     col);
      int ad1 = groupAddr<G>(r0 + 1, col);
      if (PERM) { ad0 ^= (ad0 >> 1); ad1 ^= (ad1 >> 1); }
      bRe[k].x = sRe[ad0]; bRe[k].y = sRe[ad1];
      bIm[k].x = sIm[ad0]; bIm[k].y = sIm[ad1];
    }
    v8f accRe = {}; v8f accIm = {};
    accRe = __builtin_amdgcn_wmma_f32_16x16x4_f32(false, a0, false, bRe[0], (short)0, accRe, false, false);
    accRe = __builtin_amdgcn_wmma_f32_16x16x4_f32(false, a1, false, bRe[1], (short)0, accRe, false, false);
    accRe = __builtin_amdgcn_wmma_f32_16x16x4_f32(false, a2, false, bRe[2], (short)0, accRe, false, false);
    accRe = __builtin_amdgcn_wmma_f32_16x16x4_f32(false, a3, false, bRe[3], (short)0, accRe, false, false);
    accIm = __builtin_amdgcn_wmma_f32_16x16x4_f32(false, a0, false, bIm[0], (short)0, accIm, false, false);
    accIm = __builtin_amdgcn_wmma_f32_16x16x4_f32(false, a1, false, bIm[1], (short)0, accIm, false, false);
    accIm = __builtin_amdgcn_wmma_f32_16x16x4_f32(false, a2, false, bIm[2], (short)0, accIm, false, false);
    accIm = __builtin_amdgcn_wmma_f32_16x16x4_f32(false, a3, false, bIm[3], (short)0, accIm, false, false);

#pragma unroll
    for (int v = 0; v < 8; ++v) {
      const int addr = groupAddr<G>(v + rbase, col);
      dRe[addr] = accRe[v];
      dIm[addr] = accIm[v];
    }
  }
}

__device__ __forceinline__ float ryElem(float c, float s, int a, int b) {
  // RY = [[c,-s],[s,c]]
  return (a == b) ? c : ((a == 1) ? s : -s);
}

extern "C" __global__ void __launch_bounds__(BLOCK)
vqc_kernel(const float* __restrict__ x, const float* __restrict__ freq,
           const float* __restrict__ vp, float* __restrict__ out)
{
  extern __shared__ float L[];
  const int tid = threadIdx.x;
  const int bs  = blockIdx.x;

  float* re0 = L + OFF_RE0; float* im0 = L + OFF_IM0;
  float* re1 = L + OFF_RE1; float* im1 = L + OFF_IM1;

  // ---- Encoding: per-wire v_i = RX(f x)*RY(x)*H |0> (closed form) ----
  if (tid < NQ) {
    const float xi = x[bs * NQ + tid];
    const float f  = freq[tid];
    const float c  = cosf(0.5f * xi), s = sinf(0.5f * xi);
    const float u0 = (c - s) * RSQRT2, u1 = (c + s) * RSQRT2;
    const float cp = cosf(0.5f * f * xi), sp = sinf(0.5f * f * xi);
    float* vt = L + OFF_VT + 4 * tid;
    vt[0] = cp * u0; vt[1] = -sp * u1;   // v[0] = cp*u0 - i*sp*u1
    vt[2] = cp * u1; vt[3] = -sp * u0;   // v[1] = cp*u1 - i*sp*u0
  }
  __syncthreads();

  // Prefix-product tables in (currently unused) buffer-1 space.
  float* Thr = re1;        float* Thi = im1;        // 256 (wires 0-7)
  float* Tlr = re1 + 256;  float* Tli = im1 + 256;  // 64  (wires 8-13)
  {
    float pr = 1.f, pi = 0.f;
#pragma unroll
    for (int q = 0; q < 8; ++q) {
      const int bit = (tid >> (7 - q)) & 1;
      const float* v = L + OFF_VT + 4 * q + 2 * bit;
      const float nr = pr * v[0] - pi * v[1];
      const float ni = pr * v[1] + pi * v[0];
      pr = nr; pi = ni;
    }
    Thr[tid] = pr; Thi[tid] = pi;
    if (tid < 64) {
      float qr = 1.f, qi = 0.f;
#pragma unroll
      for (int q = 0; q < 6; ++q) {
        const int bit = (tid >> (5 - q)) & 1;
        const float* v = L + OFF_VT + 4 * (8 + q) + 2 * bit;
        const float nr = qr * v[0] - qi * v[1];
        const float ni = qr * v[1] + qi * v[0];
        qr = nr; qi = ni;
      }
      Tlr[tid] = qr; Tli[tid] = qi;
    }
  }
  __syncthreads();

  for (int idx = tid; idx < DIM; idx += BLOCK) {
    const int hi = idx >> 6, lo = idx & 63;
    const float ar = Thr[hi], ai = Thi[hi], br = Tlr[lo], bi = Tli[lo];
    re0[idx] = ar * br - ai * bi;
    im0[idx] = ar * bi + ai * br;
  }
  __syncthreads();

  // ---- Variational layers: Gray-code permutation + RY Kronecker GEMMs ----
  for (int layer = 0; layer < VDEPTH; ++layer) {
    float* sRe = (layer & 1) ? re1 : re0;
    float* sIm = (layer & 1) ? im1 : im0;
    float* dRe = (layer & 1) ? re0 : re1;
    float* dIm = (layer & 1) ? im0 : im1;

    if (tid < NQ) {
      const float th = vp[layer * NQ + tid];
      L[OFF_CS + 2 * tid]     = cosf(0.5f * th);
      L[OFF_CS + 2 * tid + 1] = sinf(0.5f * th);
    }
    __syncthreads();

    // Build three 16x16 Kronecker RY matrices (wires 4g..4g+3).
    for (int e = tid; e < 768; e += BLOCK) {
      const int g = e >> 8, ii = (e >> 4) & 15, jc = e & 15;
      float p = 1.f;
#pragma unroll
      for (int q = 0; q < 4; ++q) {
        const int w = 4 * g + q;
        const int a = (ii >> (3 - q)) & 1, b = (jc >> (3 - q)) & 1;
        p *= ryElem(L[OFF_CS + 2 * w], L[OFF_CS + 2 * w + 1], a, b);
      }
      L[OFF_M + e] = p;
    }
    if (tid < 16) {  // 4x4 for wires 12 (bit1) and 13 (bit0)
      const int rr = tid >> 2, cc = tid & 3;
      float p = ryElem(L[OFF_CS + 24], L[OFF_CS + 25], rr >> 1, cc >> 1);
      p      *= ryElem(L[OFF_CS + 26], L[OFF_CS + 27], rr & 1,  cc & 1);
      L[OFF_M3 + tid] = p;
    }
    __syncthreads();

    // Group 0 with fused CNOT-ladder permutation (src -> dst buffer).
    apply_group16<0, true >(sRe, sIm, dRe, dIm, L + OFF_M);
    __syncthreads();
    apply_group16<1, false>(dRe, dIm, dRe, dIm, L + OFF_M + 256);
    __syncthreads();
    apply_group16<2, false>(dRe, dIm, dRe, dIm, L + OFF_M + 512);
    __syncthreads();

    // Wires 12-13: 4x4 real matvec on contiguous 4-element fibers, in place.
    const float* M3 = L + OFF_M3;
    for (int f = tid; f < DIM / 4; f += BLOCK) {
      const int base = f << 2;
      float ir[4], ii4[4], orr[4], oii[4];
#pragma unroll
      for (int q = 0; q < 4; ++q) { ir[q] = dRe[base + q]; ii4[q] = dIm[base + q]; }
#pragma unroll
      for (int a = 0; a < 4; ++a) {
        float sr = 0.f, si = 0.f;
#pragma unroll
        for (int b = 0; b < 4; ++b) { sr += M3[a * 4 + b] * ir[b]; si += M3[a * 4 + b] * ii4[b]; }
        orr[a] = sr; oii[a] = si;
      }
#pragma unroll
      for (int q = 0; q < 4; ++q) { dRe[base + q] = orr[q]; dIm[base + q] = oii[q]; }
    }
    __syncthreads();
  }

  // After 6 layers the state lives in buffer 0.  <Z_0>: sign from bit 13.
  float acc = 0.f;
  for (int idx = tid; idx < DIM; idx += BLOCK) {
    const float p = re0[idx] * re0[idx] + im0[idx] * im0[idx];
    acc += (idx & 0x2000) ? -p : p;
  }
  float* red = L + OFF_M;
  red[tid] = acc;
  __syncthreads();
  for (int s = BLOCK / 2; s > 0; s >>= 1) {
    if (tid < s) red[tid] += red[tid + s];
    __syncthreads();
  }
  if (tid == 0) out[bs] = red[0];
}

extern "C" void kernel_launch(void* const* d_in, const int* in_sizes, int n_in,
                              void* d_out, int out_size, void* d_ws, size_t ws_size,
                              hipStream_t stream) {
  const float* x    = (const float*)d_in[0];   // [B, 14]
  const float* freq = (const float*)d_in[1];   // [14]
  const float* vp   = (const float*)d_in[2];   // [6, 14]
  float* out        = (float*)d_out;           // [B, 1]
  const int B = in_sizes[0] / NQ;
  const size_t shmem = (size_t)LDS_FLOATS * sizeof(float);
  vqc_kernel<<<B, BLOCK, shmem, stream>>>(x, freq, vp, out);
}